// PostProcessing3_3590592659710
// MI455X (gfx1250) — compile-verified
//
#include <hip/hip_runtime.h>
#include <hip/hip_bf16.h>
#include <math.h>

#define HH 512
#define WW 1024
#define HWPX 524288          // HH*WW = 2^19
#define NCLS 34
#define KTOP 50
#define KPAD 64
#define NTH 10               // thing classes 24..33
#define MCODE 500
#define CAP 16384
#define INT_MAX_ 0x7fffffff
#define INT_MIN_ (-0x7fffffff-1)

typedef float v2f __attribute__((ext_vector_type(2)));
typedef float v8f __attribute__((ext_vector_type(8)));

#ifndef __has_builtin
#define __has_builtin(x) 0
#endif
#if __has_builtin(__builtin_amdgcn_wmma_f32_16x16x4_f32)
#define USE_WMMA_X4 1
#else
#define USE_WMMA_X4 0
#endif

// ---------------- per-pixel argmax + softmax stats for last batch ----------------
__global__ __launch_bounds__(256)
void k_segmap(const float* __restrict__ logits, int* __restrict__ segcls,
              float* __restrict__ out_segmap, float* __restrict__ mx,
              float* __restrict__ se, int B) {
  int p = blockIdx.x * 256 + threadIdx.x;
  int b = p >> 19, pp = p & (HWPX - 1);
  const float* lp = logits + (size_t)b * NCLS * HWPX + pp;
  float vals[NCLS];
  vals[0] = lp[0];
  float best = vals[0];
  int bc = 0;
#pragma unroll
  for (int c = 1; c < NCLS; ++c) {
    float v = lp[(size_t)c * HWPX];
    vals[c] = v;
    if (v > best) { best = v; bc = c; }
  }
  segcls[p] = bc;
  out_segmap[p] = (float)bc;
  if (b == B - 1) {
    float s = 0.f;
#pragma unroll
    for (int c = 0; c < NCLS; ++c) s += expf(vals[c] - best);
    mx[pp] = best;
    se[pp] = s;
  }
}

// ---------------- vote scatter (round half-to-even like jnp.round) ----------------
__global__ __launch_bounds__(256)
void k_vote(const int* __restrict__ segcls, const float* __restrict__ regs,
            int* __restrict__ vote, int class_i) {
  int p = blockIdx.x * 256 + threadIdx.x;
  if (segcls[p] != class_i) return;
  int b = p >> 19, pp = p & (HWPX - 1);
  float pr = (float)((pp >> 10) + 1) - regs[(size_t)(b * 2 + 1) * HWPX + pp];
  float pc = (float)((pp & 1023) + 1) - regs[(size_t)(b * 2 + 0) * HWPX + pp];
  int r = __float2int_rn(pr), c = __float2int_rn(pc);
  if (r >= 0 && c >= 0 && r < HH && c < WW)
    atomicAdd(&vote[(b << 19) + (r << 10) + c], 1);
}

// ---------------- fused 7x7 maxpool NMS + threshold(>10) + candidate compaction ----
__global__ __launch_bounds__(256)
void k_nmscand(const int* __restrict__ vote, int* __restrict__ candIdx,
               int* __restrict__ candVal, int* __restrict__ candCnt) {
  int p = blockIdx.x * 256 + threadIdx.x;
  int b = p >> 19, pp = p & (HWPX - 1);
  int v = vote[p];
  if (v <= 10) return;
  int row = pp >> 10, col = pp & 1023;
  int r0 = max(row - 3, 0), r1 = min(row + 3, HH - 1);
  int c0 = max(col - 3, 0), c1 = min(col + 3, WW - 1);
  bool peak = true;
  for (int r = r0; r <= r1 && peak; ++r)
    for (int c = c0; c <= c1; ++c)
      if (vote[(b << 19) + (r << 10) + c] > v) { peak = false; break; }
  if (!peak) return;
  int slot = atomicAdd(&candCnt[b], 1);
  if (slot < CAP) { candIdx[b * CAP + slot] = pp; candVal[b * CAP + slot] = v; }
}

// ---------------- top-K selection: (max value, min index) K times ----------------
__global__ __launch_bounds__(256)
void k_topk(const int* __restrict__ candIdx, int* __restrict__ candVal,
            const int* __restrict__ candCnt, float* __restrict__ crA,
            float* __restrict__ ccA, unsigned long long* __restrict__ maskHas,
            int* __restrict__ flag) {
  int b = blockIdx.x, t = threadIdx.x;
  __shared__ int sbv[256], sbi[256], sbs[256];
  if (t < KPAD) { crA[b * KPAD + t] = 0.f; ccA[b * KPAD + t] = 0.f; }
  int n = candCnt[b]; if (n > CAP) n = CAP;
  unsigned long long km = 0ull;
  for (int k = 0; k < KTOP; ++k) {
    int bv = -1, bi = INT_MAX_, bs = -1;
    for (int i = t; i < n; i += 256) {
      int v = candVal[b * CAP + i], id = candIdx[b * CAP + i];
      bool take = (v > bv) | ((v == bv) & (id < bi));
      bv = take ? v : bv; bi = take ? id : bi; bs = take ? i : bs;
    }
    sbv[t] = bv; sbi[t] = bi; sbs[t] = bs;
    __syncthreads();
    for (int s = 128; s > 0; s >>= 1) {
      if (t < s) {
        bool take = (sbv[t + s] > sbv[t]) |
                    ((sbv[t + s] == sbv[t]) & (sbi[t + s] < sbi[t]));
        if (take) { sbv[t] = sbv[t + s]; sbi[t] = sbi[t + s]; sbs[t] = sbs[t + s]; }
      }
      __syncthreads();
    }
    if (t == 0 && sbv[0] > 0) {
      km |= (1ull << k);
      crA[b * KPAD + k] = (float)(sbi[0] >> 10);
      ccA[b * KPAD + k] = (float)(sbi[0] & 1023);
      candVal[b * CAP + sbs[0]] = -1;  // consume
    }
    __syncthreads();
  }
  if (t == 0) { maskHas[b] = km; flag[b] = 1; }
}

// ---------------- WMMA clustering: argmin_k over pairwise distances ----------------
// score[p][k] = -2*pr*cr - 2*pc*cc + (cr^2+cc^2)  == d2 - (pr^2+pc^2); same argmin.
// A (16x4, f32): rows = pixels (pr, pc, 1, 0); B (4x16): columns = centers.
// Argmin uses a monotone-packed u64 key (orderedFloat<<32 | centerIdx): a single
// branch-free u64 min reproduces jnp.argmin first-min tie-breaking exactly.

__device__ __forceinline__ unsigned long long pack_key(float v, int n) {
  unsigned int ob = __float_as_uint(v);
  ob ^= ((unsigned int)((int)ob >> 31)) | 0x80000000u;  // monotone float->uint
  return ((unsigned long long)ob << 32) | (unsigned int)n;
}

__device__ __forceinline__ unsigned long long kmin(unsigned long long a,
                                                   unsigned long long b) {
  return b < a ? b : a;
}

// 16-lane XOR min-reduce (halves stay independent for XOR offsets <= 8).
// ds_swizzle imm: group-of-32 mode, new_lane = lane ^ off  ->  (off<<10) | 0x1f.
#define SWZ_STEP(key, IMM)                                                        \
  {                                                                               \
    unsigned int lo = (unsigned int)(key), hi = (unsigned int)((key) >> 32);      \
    unsigned int olo = (unsigned int)__builtin_amdgcn_ds_swizzle((int)lo, IMM);   \
    unsigned int ohi = (unsigned int)__builtin_amdgcn_ds_swizzle((int)hi, IMM);   \
    unsigned long long ok = ((unsigned long long)ohi << 32) | olo;                \
    key = kmin(key, ok);                                                          \
  }

__global__ __launch_bounds__(256)
void k_cluster(const int* __restrict__ segcls, const float* __restrict__ regs,
               const float* __restrict__ crA, const float* __restrict__ ccA,
               const unsigned long long* __restrict__ maskP,
               const int* __restrict__ flagP, int class_i, int* __restrict__ inst) {
  const int wave = threadIdx.x >> 5, lane = threadIdx.x & 31;
  const int m = lane & 15, hs = lane >> 4;
  const int pbase = (blockIdx.x * 8 + wave) * 16;
  const int b = pbase >> 19, p0 = pbase & (HWPX - 1);
  if (flagP[b] == 0) return;  // uniform per block (block spans 128 pixels in one batch)
  const unsigned long long km = maskP[b];

  const int pp0 = p0 + m;
  const float pr = (float)((pp0 >> 10) + 1) - regs[(size_t)(b * 2 + 1) * HWPX + pp0];
  const float pc = (float)((pp0 & 1023) + 1) - regs[(size_t)(b * 2 + 0) * HWPX + pp0];

  float sc[4][8];  // scores: [center-group][row-within-half]

#if USE_WMMA_X4
  // A layout (16x4 f32): lanes 0-15 hold K=0,1 of row M=lane; lanes 16-31 hold K=2,3.
  v2f A; A.x = hs ? 1.0f : pr; A.y = hs ? 0.0f : pc;
#pragma unroll
  for (int g = 0; g < 4; ++g) {
    int n = (g << 4) + m;
    float cr = crA[b * KPAD + n], cc = ccA[b * KPAD + n];
    // B layout (4x16 f32): VGPR0 = rows K=0 (lanes 0-15) / K=2 (lanes 16-31), VGPR1 = K=1 / K=3.
    v2f Bv;
    Bv.x = hs ? (cr * cr + cc * cc) : (-2.0f * cr);
    Bv.y = hs ? 0.0f : (-2.0f * cc);
    v8f C = {0.f, 0.f, 0.f, 0.f, 0.f, 0.f, 0.f, 0.f};
    C = __builtin_amdgcn_wmma_f32_16x16x4_f32(false, A, false, Bv, (short)0, C,
                                              false, false);
#pragma unroll
    for (int r = 0; r < 8; ++r) sc[g][r] = C[r];
  }
#else
#pragma unroll
  for (int g = 0; g < 4; ++g) {
    int n = (g << 4) + m;
    float cr = crA[b * KPAD + n], cc = ccA[b * KPAD + n];
    float s0 = -2.0f * cr, s1 = -2.0f * cc, s2 = cr * cr + cc * cc;
#pragma unroll
    for (int r = 0; r < 8; ++r) {
      float prr = __shfl(pr, r + hs * 8, 32), pcc = __shfl(pc, r + hs * 8, 32);
      sc[g][r] = s0 * prr + s1 * pcc + s2;
    }
  }
#endif

  // D layout: row r holds pixel M=r (lanes 0-15) / M=r+8 (lanes 16-31), N=lane&15.
  unsigned long long mykey = ~0ull;
#pragma unroll
  for (int r = 0; r < 8; ++r) {
    // fold the 4 center-groups in registers first (branch-free u64 min)
    unsigned long long key = ~0ull;
#pragma unroll
    for (int g = 0; g < 4; ++g) {
      int n = (g << 4) + m;
      bool kept = (km >> n) & 1ull;
      float v = kept ? sc[g][r] : 3.0e30f;
      key = kmin(key, pack_key(v, n));
    }
    // then one 4-step 16-lane reduction (2 ds_swizzle per step, no addr VALU)
    SWZ_STEP(key, 0x041f)   // xor 1
    SWZ_STEP(key, 0x081f)   // xor 2
    SWZ_STEP(key, 0x101f)   // xor 4
    SWZ_STEP(key, 0x201f)   // xor 8
    bool take = (m == r) & (key < mykey);
    mykey = take ? key : mykey;
  }

  if (m < 8) {  // lane m of each half writes pixel p0 + hs*8 + m
    int pp = p0 + (hs << 3) + m;
    int things = (segcls[(b << 19) + pp] == class_i);
    int myN = (int)(mykey & 63ull);
    inst[(b << 19) + pp] = (things && km != 0ull) ? (myN + 1) : 0;
  }
}

// ---------------- segment statistics (bbox + regression, deterministic) -----------
__global__ __launch_bounds__(256)
void k_statinit(int* cnt, int* ymin, int* ymax, int* xmin, int* xmax, int* cnt2,
                unsigned long long* sx, unsigned long long* sy, int n) {
  int i = blockIdx.x * 256 + threadIdx.x;
  if (i >= n) return;
  cnt[i] = 0; cnt2[i] = 0;
  ymin[i] = INT_MAX_; ymax[i] = INT_MIN_;
  xmin[i] = INT_MAX_; xmax[i] = INT_MIN_;
  sx[i] = 0ull; sy[i] = 0ull;
}

__global__ __launch_bounds__(256)
void k_segstats(const int* __restrict__ inst, int* cnt, int* ymin, int* ymax,
                int* xmin, int* xmax) {
  int p = blockIdx.x * 256 + threadIdx.x;
  int iv = inst[p];
  if (iv <= 0) return;
  int b = p >> 19, pp = p & (HWPX - 1);
  int s = b * (KTOP + 1) + iv;
  atomicAdd(&cnt[s], 1);
  atomicMin(&ymin[s], pp >> 10);
  atomicMax(&ymax[s], pp >> 10);
  atomicMin(&xmin[s], pp & 1023);
  atomicMax(&xmax[s], pp & 1023);
}

__global__ __launch_bounds__(256)
void k_regstats(const int* __restrict__ inst, const float* __restrict__ regs,
                int* cnt2, unsigned long long* sx, unsigned long long* sy) {
  int p = blockIdx.x * 256 + threadIdx.x;
  int iv = inst[p];
  if (iv <= 0) return;
  int b = p >> 19, pp = p & (HWPX - 1);
  int s = b * (KTOP + 1) + iv;
  atomicAdd(&cnt2[s], 1);
  float rx = regs[(size_t)(b * 2 + 0) * HWPX + pp];
  float ry = regs[(size_t)(b * 2 + 1) * HWPX + pp];
  atomicAdd(&sx[s], (unsigned long long)(long long)llrintf(rx * 65536.0f));
  atomicAdd(&sy[s], (unsigned long long)(long long)llrintf(ry * 65536.0f));
}

// ---------------- pruning decisions (single block) --------------------------------
__global__ __launch_bounds__(128)
void k_decide_bbox(const int* cnt, const int* ymin, const int* ymax, const int* xmin,
                   const int* xmax, const unsigned long long* maskHas,
                   unsigned long long* maskCur, int* flag, int B) {
  __shared__ unsigned int glo[8], ghi[8];
  int t = threadIdx.x;
  if (t < B) { glo[t] = 0u; ghi[t] = 0u; }
  __syncthreads();
  if (t < B * KTOP) {
    int b = t / KTOP, k = t % KTOP + 1, s = b * (KTOP + 1) + k;
    int c = cnt[s];
    if (c > 0) {
      int area = (ymax[s] - ymin[s] + 1) * (xmax[s] - xmin[s] + 1);
      if ((float)c / (float)area > 0.3f) {
        int bit = k - 1;
        if (bit < 32) atomicOr(&glo[b], 1u << bit);
        else atomicOr(&ghi[b], 1u << (bit - 32));
      }
    }
  }
  __syncthreads();
  if (t < B) {
    unsigned long long g = ((unsigned long long)ghi[t] << 32) | glo[t];
    maskCur[t] = g & maskHas[t];
    flag[t] = g ? 1 : 0;
  }
}

__global__ __launch_bounds__(128)
void k_decide_reg(const int* cnt2, const unsigned long long* sx,
                  const unsigned long long* sy, const unsigned long long* maskHas,
                  unsigned long long* maskCur, int* flag, int B) {
  __shared__ unsigned int glo[8], ghi[8];
  int t = threadIdx.x;
  if (t < B) { glo[t] = 0u; ghi[t] = 0u; }
  __syncthreads();
  if (t < B * KTOP) {
    int b = t / KTOP, k = t % KTOP + 1, s = b * (KTOP + 1) + k;
    int c = cnt2[s];
    if (c > 0) {
      float mxr = (float)((double)(long long)sx[s] / 65536.0) / (float)c;
      float myr = (float)((double)(long long)sy[s] / 65536.0) / (float)c;
      if (fabsf(mxr) < 3.0f && fabsf(myr) < 3.0f) {
        int bit = k - 1;
        if (bit < 32) atomicOr(&glo[b], 1u << bit);
        else atomicOr(&ghi[b], 1u << (bit - 32));
      }
    }
  }
  __syncthreads();
  if (t < B) {
    unsigned long long g = ((unsigned long long)ghi[t] << 32) | glo[t];
    maskCur[t] = g & maskHas[t];
    flag[t] = g ? 1 : 0;
  }
}

// ---------------- code accumulation, relabel, histogram, probs --------------------
__global__ __launch_bounds__(256)
void k_accum(const int* __restrict__ inst, int* __restrict__ total, int ci) {
  int p = blockIdx.x * 256 + threadIdx.x;
  int iv = inst[p];
  if (iv > 0) total[p] += ci * KTOP + iv;
}

__global__ __launch_bounds__(256)
void k_pres(const int* __restrict__ total, int* __restrict__ pres) {
  int p = blockIdx.x * 256 + threadIdx.x;
  int b = p >> 19;
  pres[b * (MCODE + 1) + total[p]] = 1;
}

__global__ void k_ranks(const int* __restrict__ pres, int* __restrict__ ranks) {
  if (threadIdx.x != 0) return;
  int b = blockIdx.x, run = 0;
  ranks[b * (MCODE + 1)] = 0;  // pres[:,0] forced to 0
  for (int m = 1; m <= MCODE; ++m) {
    run += pres[b * (MCODE + 1) + m] ? 1 : 0;
    ranks[b * (MCODE + 1) + m] = run;
  }
}

__global__ __launch_bounds__(256)
void k_color(const int* __restrict__ total, const int* __restrict__ ranks,
             const int* __restrict__ segcls, float* __restrict__ out_colored,
             int* __restrict__ colored_i, int* __restrict__ hist, int B) {
  int p = blockIdx.x * 256 + threadIdx.x;
  int b = p >> 19, pp = p & (HWPX - 1);
  int col = ranks[b * (MCODE + 1) + total[p]];
  out_colored[p] = (float)col;
  colored_i[p] = col;
  int cl = segcls[(size_t)(B - 1) * HWPX + pp];  // source bug: last batch's seg map
  atomicAdd(&hist[(b * (MCODE + 1) + col) * NCLS + cl], 1);
}

__global__ __launch_bounds__(256)
void k_instcls(const int* __restrict__ hist, int* __restrict__ instcls,
               int* __restrict__ cnts, float* __restrict__ out_ic,
               float* __restrict__ out_cnt, int B) {
  int i = blockIdx.x * 256 + threadIdx.x;
  if (i >= B * (MCODE + 1)) return;
  const int* h = hist + (size_t)i * NCLS;
  int bc = 0, bv = h[0], s = h[0];
#pragma unroll
  for (int c = 1; c < NCLS; ++c) {
    int v = h[c];
    s += v;
    if (v > bv) { bv = v; bc = c; }
  }
  instcls[i] = bc;
  cnts[i] = s;
  int b = i / (MCODE + 1), m = i % (MCODE + 1);
  if (m >= 1) {
    out_ic[b * MCODE + m - 1] = (float)bc;
    out_cnt[b * MCODE + m - 1] = (float)s;
  }
}

__global__ __launch_bounds__(256)
void k_prob(const float* __restrict__ logits, const float* __restrict__ mx,
            const float* __restrict__ se, const int* __restrict__ colored_i,
            const int* __restrict__ instcls, unsigned long long* __restrict__ psum,
            int B) {
  int p = blockIdx.x * 256 + threadIdx.x;
  int b = p >> 19, pp = p & (HWPX - 1);
  int col = colored_i[p];
  int cls = instcls[b * (MCODE + 1) + col];
  float lg = logits[((size_t)(B - 1) * NCLS + cls) * HWPX + pp];
  float pr = expf(lg - mx[pp]) / se[pp];
  atomicAdd(&psum[b * (MCODE + 1) + col],
            (unsigned long long)(long long)llrintf(pr * 1048576.0f));
}

__global__ __launch_bounds__(256)
void k_final(const unsigned long long* __restrict__ psum, const int* __restrict__ cnts,
             float* __restrict__ out_prob, int B) {
  int i = blockIdx.x * 256 + threadIdx.x;
  if (i >= B * MCODE) return;
  int b = i / MCODE, m = i % MCODE + 1;
  float pv = (float)((double)psum[b * (MCODE + 1) + m] * (1.0 / 1048576.0));
  int c = cnts[b * (MCODE + 1) + m];
  out_prob[i] = pv / (float)(c > 0 ? c : 1);
}

// =================================================================================
extern "C" void kernel_launch(void* const* d_in, const int* in_sizes, int n_in,
                              void* d_out, int out_size, void* d_ws, size_t ws_size,
                              hipStream_t stream) {
  const float* logits = (const float*)d_in[0];
  const float* regs = (const float*)d_in[2];
  const int B = in_sizes[0] / (NCLS * HWPX);
  const size_t BHW = (size_t)B * HWPX;
  const int NS = B * (KTOP + 1);

  char* ws = (char*)d_ws;
  size_t off = 0;
  auto alloc = [&](size_t bytes) -> void* {
    void* p = ws + off;
    off = (off + bytes + 255) & ~(size_t)255;
    return p;
  };
  int* w_segcls = (int*)alloc(BHW * 4);
  int* w_total = (int*)alloc(BHW * 4);
  int* w_inst = (int*)alloc(BHW * 4);
  int* w_vote = (int*)alloc(BHW * 4);       // reused as colored_i after class loop
  float* w_mx = (float*)alloc((size_t)HWPX * 4);
  float* w_se = (float*)alloc((size_t)HWPX * 4);
  int* w_cidx = (int*)alloc((size_t)B * CAP * 4);
  int* w_cval = (int*)alloc((size_t)B * CAP * 4);
  int* w_ccnt = (int*)alloc(B * 4);
  float* w_cr = (float*)alloc((size_t)B * KPAD * 4);
  float* w_cc = (float*)alloc((size_t)B * KPAD * 4);
  unsigned long long* w_mHas = (unsigned long long*)alloc(B * 8);
  unsigned long long* w_mCur = (unsigned long long*)alloc(B * 8);
  int* w_flag = (int*)alloc(B * 4);
  int* w_cnt = (int*)alloc(NS * 4);
  int* w_ymin = (int*)alloc(NS * 4);
  int* w_ymax = (int*)alloc(NS * 4);
  int* w_xmin = (int*)alloc(NS * 4);
  int* w_xmax = (int*)alloc(NS * 4);
  int* w_cnt2 = (int*)alloc(NS * 4);
  unsigned long long* w_sx = (unsigned long long*)alloc(NS * 8);
  unsigned long long* w_sy = (unsigned long long*)alloc(NS * 8);
  int* w_pres = (int*)alloc((size_t)B * (MCODE + 1) * 4);
  int* w_ranks = (int*)alloc((size_t)B * (MCODE + 1) * 4);
  int* w_hist = (int*)alloc((size_t)B * (MCODE + 1) * NCLS * 4);
  int* w_icls = (int*)alloc((size_t)B * (MCODE + 1) * 4);
  int* w_cnts = (int*)alloc((size_t)B * (MCODE + 1) * 4);
  unsigned long long* w_psum = (unsigned long long*)alloc((size_t)B * (MCODE + 1) * 8);

  float* o = (float*)d_out;
  float* o_colored = o;
  float* o_segmap = o + BHW;
  float* o_ic = o + 2 * BHW;
  float* o_prob = o + 2 * BHW + (size_t)B * MCODE;
  float* o_cnt = o + 2 * BHW + 2 * (size_t)B * MCODE;

  dim3 blk(256);
  dim3 gpix((unsigned)(BHW / 256));
  dim3 gclu((unsigned)(BHW / 128));

  k_segmap<<<gpix, blk, 0, stream>>>(logits, w_segcls, o_segmap, w_mx, w_se, B);
  hipMemsetAsync(w_total, 0, BHW * 4, stream);

  for (int ci = 0; ci < NTH; ++ci) {
    const int class_i = 24 + ci;
    hipMemsetAsync(w_vote, 0, BHW * 4, stream);
    hipMemsetAsync(w_ccnt, 0, B * 4, stream);
    k_vote<<<gpix, blk, 0, stream>>>(w_segcls, regs, w_vote, class_i);
    k_nmscand<<<gpix, blk, 0, stream>>>(w_vote, w_cidx, w_cval, w_ccnt);
    k_topk<<<dim3(B), blk, 0, stream>>>(w_cidx, w_cval, w_ccnt, w_cr, w_cc, w_mHas,
                                        w_flag);
    // pass 1: keep = has_c (always runs)
    k_cluster<<<gclu, blk, 0, stream>>>(w_segcls, regs, w_cr, w_cc, w_mHas, w_flag,
                                        class_i, w_inst);
    // bbox pruning
    k_statinit<<<dim3(1), blk, 0, stream>>>(w_cnt, w_ymin, w_ymax, w_xmin, w_xmax,
                                            w_cnt2, w_sx, w_sy, NS);
    k_segstats<<<gpix, blk, 0, stream>>>(w_inst, w_cnt, w_ymin, w_ymax, w_xmin, w_xmax);
    k_decide_bbox<<<dim3(1), dim3(128), 0, stream>>>(w_cnt, w_ymin, w_ymax, w_xmin,
                                                     w_xmax, w_mHas, w_mCur, w_flag, B);
    k_cluster<<<gclu, blk, 0, stream>>>(w_segcls, regs, w_cr, w_cc, w_mCur, w_flag,
                                        class_i, w_inst);
    // regression pruning
    k_statinit<<<dim3(1), blk, 0, stream>>>(w_cnt, w_ymin, w_ymax, w_xmin, w_xmax,
                                            w_cnt2, w_sx, w_sy, NS);
    k_regstats<<<gpix, blk, 0, stream>>>(w_inst, regs, w_cnt2, w_sx, w_sy);
    k_decide_reg<<<dim3(1), dim3(128), 0, stream>>>(w_cnt2, w_sx, w_sy, w_mHas, w_mCur,
                                                    w_flag, B);
    k_cluster<<<gclu, blk, 0, stream>>>(w_segcls, regs, w_cr, w_cc, w_mCur, w_flag,
                                        class_i, w_inst);
    k_accum<<<gpix, blk, 0, stream>>>(w_inst, w_total, ci);
  }

  // relabel present codes -> consecutive ranks
  hipMemsetAsync(w_pres, 0, (size_t)B * (MCODE + 1) * 4, stream);
  hipMemsetAsync(w_hist, 0, (size_t)B * (MCODE + 1) * NCLS * 4, stream);
  hipMemsetAsync(w_psum, 0, (size_t)B * (MCODE + 1) * 8, stream);
  k_pres<<<gpix, blk, 0, stream>>>(w_total, w_pres);
  k_ranks<<<dim3(B), dim3(32), 0, stream>>>(w_pres, w_ranks);
  k_color<<<gpix, blk, 0, stream>>>(w_total, w_ranks, w_segcls, o_colored, w_vote,
                                    w_hist, B);
  k_instcls<<<dim3((B * (MCODE + 1) + 255) / 256), blk, 0, stream>>>(
      w_hist, w_icls, w_cnts, o_ic, o_cnt, B);
  k_prob<<<gpix, blk, 0, stream>>>(logits, w_mx, w_se, w_vote, w_icls, w_psum, B);
  k_final<<<dim3((B * MCODE + 255) / 256), blk, 0, stream>>>(w_psum, w_cnts, o_prob, B);
}